// LinearAttention_5892695130487
// MI455X (gfx1250) — compile-verified
//
#include <hip/hip_runtime.h>
#include <math.h>

typedef __attribute__((ext_vector_type(2))) float v2f;
typedef __attribute__((ext_vector_type(8))) float v8f;
typedef int v4i __attribute__((vector_size(16)));

#define B_ 16
#define N_ 8192
#define D_ 128
#define H_ 64
#define NSEG 16
#define NCHUNK (N_ / NSEG)   // 512 rows per KV segment
#define BR 32                // rows staged per LDS block in kv kernel

#define AS1 __attribute__((address_space(1)))
#define AS3 __attribute__((address_space(3)))

static __device__ __forceinline__ v8f v8f_zero() {
  v8f z = {0.f, 0.f, 0.f, 0.f, 0.f, 0.f, 0.f, 0.f};
  return z;
}

// D = A(16x4 f32) * B(4x16 f32) + C(16x16 f32), wave32
static __device__ __forceinline__ v8f wmma_f32_16x16x4(v2f a, v2f b, v8f c) {
  return __builtin_amdgcn_wmma_f32_16x16x4_f32(
      /*neg_a=*/false, a, /*neg_b=*/false, b,
      /*c_mod=*/(short)0, c, /*reuse_a=*/false, /*reuse_b=*/false);
}

static __device__ __forceinline__ float gelu_exact(float v) {
  return 0.5f * v * (1.0f + erff(v * 0.70710678118654752f));
}

// 16-byte global -> LDS async copy (ASYNCcnt-tracked DMA)
static __device__ __forceinline__ void async_copy_b128(const float* g, float* l) {
#if __has_builtin(__builtin_amdgcn_global_load_async_to_lds_b128)
  __builtin_amdgcn_global_load_async_to_lds_b128(
      (AS1 v4i*)g, (AS3 v4i*)l, 0, 0);
#else
  unsigned ldsaddr = (unsigned)(unsigned long)(AS3 void*)l;
  asm volatile("global_load_async_to_lds_b128 %0, %1, off"
               :: "v"(ldsaddr), "v"(g) : "memory");
#endif
}

static __device__ __forceinline__ void async_wait_all() {
#if __has_builtin(__builtin_amdgcn_s_wait_asynccnt)
  __builtin_amdgcn_s_wait_asynccnt(0);
#else
  asm volatile("s_wait_asynccnt 0x0" ::: "memory");
#endif
}

static __device__ __forceinline__ v2f lds_pair(const float* p) {
  return *(const v2f*)p;
}

// ---------------- Kernel 1: scorer  scores = (gelu(x@w1+b1))@w2 + b2 -------
__global__ __launch_bounds__(256) void scorer_kernel(
    const float* __restrict__ x, const float* __restrict__ w1,
    const float* __restrict__ b1, const float* __restrict__ w2,
    const float* __restrict__ b2, float* __restrict__ scores) {
  __shared__ float w1P[D_ / 2][H_][2];  // 32 KB, row-pair interleaved [k/2][h][k&1]
  __shared__ float hL[8][16][16];       // 8 KB, per-wave gelu(h) tile

  const int tid  = threadIdx.x;
  const int lane = tid & 31;
  const int wave = tid >> 5;
  const int m    = lane & 15;
  const int kh   = lane >> 4;

  for (int i = tid; i < D_ * H_; i += 256) {
    const int k = i >> 6, h = i & (H_ - 1);
    w1P[k >> 1][h][k & 1] = w1[i];
  }
  __syncthreads();

  const size_t rowBase = (size_t)blockIdx.x * 128 + (size_t)wave * 16;

  v8f acc[4];
#pragma unroll
  for (int t = 0; t < 4; ++t) acc[t] = v8f_zero();

  const float* xrow = x + (rowBase + (size_t)m) * D_;  // A row for this lane
#pragma unroll 4
  for (int kb = 0; kb < D_; kb += 4) {
    v2f a = lds_pair(xrow + kb + 2 * kh);  // contiguous global b64
#pragma unroll
    for (int t = 0; t < 4; ++t) {
      v2f bf = lds_pair(&w1P[(kb >> 1) + kh][t * 16 + m][0]);  // single ds b64
      acc[t] = wmma_f32_16x16x4(a, bf, acc[t]);
    }
  }

  float scoreReg = 0.0f;
  for (int t = 0; t < 4; ++t) {
#pragma unroll
    for (int v = 0; v < 8; ++v) {
      const float h = acc[t][v] + b1[t * 16 + m];
      hL[wave][v + 8 * kh][m] = gelu_exact(h);
    }
    __syncthreads();
    if (lane < 16) {
#pragma unroll
      for (int c = 0; c < 16; ++c)
        scoreReg += hL[wave][lane][c] * w2[t * 16 + c];
    }
    __syncthreads();
  }
  if (lane < 16) scores[rowBase + lane] = scoreReg + b2[0];
}

// ---------------- Kernel 2: adassmax softmax over N per batch --------------
__global__ __launch_bounds__(1024) void softmax_kernel(
    const float* __restrict__ scores, const float* __restrict__ log_alpha,
    float* __restrict__ weights) {
  __shared__ float red[1024];
  const int b = blockIdx.x;
  const int tid = threadIdx.x;

  const float laf = log_alpha[0];
  const float alpha = fminf(fmaxf(log1pf(expf(laf)) + 1.0f, 1.0f), 10.0f);
  const float scale = alpha * rsqrtf((float)N_);

  const float* s = scores + (size_t)b * N_;
  float t[N_ / 1024];
  float lmax = -INFINITY;
#pragma unroll
  for (int i = 0; i < N_ / 1024; ++i) {
    t[i] = s[tid + i * 1024] * scale;
    lmax = fmaxf(lmax, t[i]);
  }
  red[tid] = lmax;
  __syncthreads();
  for (int st = 512; st > 0; st >>= 1) {
    if (tid < st) red[tid] = fmaxf(red[tid], red[tid + st]);
    __syncthreads();
  }
  const float mx = red[0];
  __syncthreads();

  float lsum = 0.f;
#pragma unroll
  for (int i = 0; i < N_ / 1024; ++i) {
    t[i] = expf(t[i] - mx);
    lsum += t[i];
  }
  red[tid] = lsum;
  __syncthreads();
  for (int st = 512; st > 0; st >>= 1) {
    if (tid < st) red[tid] += red[tid + st];
    __syncthreads();
  }
  const float inv = 1.0f / (red[0] + 1e-8f);
  float* w = weights + (size_t)b * N_;
#pragma unroll
  for (int i = 0; i < N_ / 1024; ++i) w[tid + i * 1024] = t[i] * inv;
}

// ---------------- Kernel 3: per-segment KV partials + S partials -----------
// KV[b,d,e] = sum_n feat[b,n,d] * xw[b,n,e];  S[b,d] = sum_n feat[b,n,d]
__global__ __launch_bounds__(256) void kv_kernel(
    const float* __restrict__ x, const float* __restrict__ weights,
    float* __restrict__ kvPart, float* __restrict__ sPart) {
  __shared__ float featP[BR / 2][D_][2];  // 16 KB, [n/2][d][n&1]
  __shared__ float xwP[BR / 2][D_][2];    // 16 KB

  const int seg = blockIdx.x;
  const int b   = blockIdx.y;
  const int tid = threadIdx.x;
  const int lane = tid & 31;
  const int wave = tid >> 5;
  const int m  = lane & 15;
  const int kh = lane >> 4;
  const int d0 = wave * 16;  // each wave owns a 16-wide d stripe

  v8f acc[8];
#pragma unroll
  for (int e = 0; e < 8; ++e) acc[e] = v8f_zero();
  float sReg = 0.0f;

  for (int blk = 0; blk < NCHUNK / BR; ++blk) {
    const int nbase = seg * NCHUNK + blk * BR;
    for (int i = tid; i < BR * D_; i += 256) {
      const int r = i >> 7, c = i & (D_ - 1);
      const size_t n = (size_t)b * N_ + nbase + r;
      const float xw = x[n * D_ + c] * weights[n];
      xwP[r >> 1][c][r & 1]   = xw;
      featP[r >> 1][c][r & 1] = (xw > 0.0f) ? (xw + 1.0f) : expf(xw);  // elu+1
    }
    __syncthreads();

    if (tid < D_) {
#pragma unroll
      for (int r = 0; r < BR; ++r) sReg += featP[r >> 1][tid][r & 1];
    }

#pragma unroll
    for (int kk = 0; kk < BR; kk += 4) {
      // A = feat^T : M=d, K=n — single ds b64 per lane
      v2f a = lds_pair(&featP[(kk >> 1) + kh][d0 + m][0]);
#pragma unroll
      for (int e = 0; e < 8; ++e) {
        v2f bf = lds_pair(&xwP[(kk >> 1) + kh][e * 16 + m][0]);
        acc[e] = wmma_f32_16x16x4(a, bf, acc[e]);
      }
    }
    __syncthreads();
  }

  float* kvp = kvPart + ((size_t)seg * B_ + b) * (D_ * D_);
#pragma unroll
  for (int e = 0; e < 8; ++e) {
#pragma unroll
    for (int v = 0; v < 8; ++v) {
      const int drow = d0 + v + 8 * kh;
      kvp[(size_t)drow * D_ + e * 16 + m] = acc[e][v];
    }
  }
  if (tid < D_) sPart[((size_t)seg * B_ + b) * D_ + tid] = sReg;
}

// ---------------- Kernel 4: deterministic partial reduction ----------------
// Final KV is written ROW-PAIR INTERLEAVED: KVp[b][d/2][e][d&1]
__global__ __launch_bounds__(256) void reduce_kernel(
    const float* __restrict__ kvPart, const float* __restrict__ sPart,
    float* __restrict__ KVp, float* __restrict__ S) {
  const size_t i = (size_t)blockIdx.x * 256 + threadIdx.x;
  const size_t nKV = (size_t)B_ * D_ * D_;
  if (i < nKV) {
    float acc = 0.f;
#pragma unroll
    for (int s = 0; s < NSEG; ++s) acc += kvPart[(size_t)s * nKV + i];
    const size_t bb = i >> 14;                 // / (D_*D_)
    const int r = (int)(i & (D_ * D_ - 1));
    const int d = r >> 7, e = r & (D_ - 1);
    KVp[bb * D_ * D_ + (((size_t)(d >> 1) * D_ + e) << 1) + (d & 1)] = acc;
  } else if (i < nKV + (size_t)B_ * D_) {
    const size_t j = i - nKV;
    float acc = 0.f;
#pragma unroll
    for (int s = 0; s < NSEG; ++s) acc += sPart[(size_t)s * B_ * D_ + j];
    S[j] = acc;
  }
}

// ---------------- Kernel 5: numerator = feat@KV, divide by feat.S ----------
__global__ __launch_bounds__(128) void attend_kernel(
    const float* __restrict__ x, const float* __restrict__ weights,
    const float* __restrict__ KVp, const float* __restrict__ S,
    float* __restrict__ out) {
  __shared__ float kvP[D_ / 2][D_][2];  // 64 KB, async-DMA'd, pair-interleaved
  __shared__ float featL[4][16][D_];    // 32 KB
  __shared__ float sL[D_];
  __shared__ float denomL[4][16];

  const int tid = threadIdx.x;
  const int lane = tid & 31;
  const int wave = tid >> 5;
  const int m  = lane & 15;
  const int kh = lane >> 4;

  const int wgPerB = N_ / 64;
  const int b = blockIdx.x / wgPerB;
  const int rowBase = (blockIdx.x % wgPerB) * 64;

  // Kick off async global->LDS DMA of this batch's 64 KB KV (L2-resident).
  const float* kvsrc = KVp + (size_t)b * D_ * D_;
  float* kvdst = &kvP[0][0][0];
  for (int i = tid * 4; i < D_ * D_; i += 128 * 4)
    async_copy_b128(kvsrc + i, kvdst + i);

  if (tid < D_) sL[tid] = S[b * D_ + tid];

  // Overlap: compute feat rows while the DMA streams KV into LDS.
  for (int i = lane; i < 16 * D_; i += 32) {
    const int r = i >> 7, c = i & (D_ - 1);
    const size_t n = (size_t)b * N_ + rowBase + wave * 16 + r;
    const float xw = x[n * D_ + c] * weights[n];
    featL[wave][r][c] = (xw > 0.0f) ? (xw + 1.0f) : expf(xw);
  }
  async_wait_all();
  __syncthreads();

  if (lane < 16) {
    float d = 0.f;
#pragma unroll 8
    for (int c = 0; c < D_; ++c) d += featL[wave][lane][c] * sL[c];
    denomL[wave][lane] = d + 1e-6f;
  }
  __syncthreads();

  float* outRow = out + ((size_t)b * N_ + rowBase + wave * 16) * D_;

  for (int e = 0; e < 8; ++e) {
    v8f acc = v8f_zero();
#pragma unroll
    for (int kb = 0; kb < D_; kb += 4) {
      // A = feat : M=row, K=d — contiguous b64 in LDS
      v2f a = lds_pair(&featL[wave][m][kb + 2 * kh]);
      // B = KV : K=d, N=e — single ds b64 thanks to pair-interleave
      v2f bf = lds_pair(&kvP[(kb >> 1) + kh][e * 16 + m][0]);
      acc = wmma_f32_16x16x4(a, bf, acc);
    }
#pragma unroll
    for (int v = 0; v < 8; ++v) {
      const int r = v + 8 * kh;
      outRow[(size_t)r * D_ + e * 16 + m] = acc[v] / denomL[wave][r];
    }
  }
}

extern "C" void kernel_launch(void* const* d_in, const int* in_sizes, int n_in,
                              void* d_out, int out_size, void* d_ws, size_t ws_size,
                              hipStream_t stream) {
  const float* x   = (const float*)d_in[0];
  const float* w1  = (const float*)d_in[1];
  const float* b1  = (const float*)d_in[2];
  const float* w2  = (const float*)d_in[3];
  const float* b2  = (const float*)d_in[4];
  const float* laf = (const float*)d_in[5];

  float* out      = (float*)d_out;
  float* attended = out;                                 // [B,N,D]
  float* scores   = out + (size_t)B_ * N_ * D_;          // [B,N]
  float* weights  = scores + (size_t)B_ * N_;            // [B,N,1]

  float* ws     = (float*)d_ws;
  float* kvPart = ws;                                    // NSEG*B*D*D
  float* sPart  = kvPart + (size_t)NSEG * B_ * D_ * D_;  // NSEG*B*D
  float* KVp    = sPart + (size_t)NSEG * B_ * D_;        // B*D*D (pair-interleaved)
  float* S      = KVp + (size_t)B_ * D_ * D_;            // B*D

  scorer_kernel<<<dim3((B_ * N_) / 128), 256, 0, stream>>>(x, w1, b1, w2, b2, scores);
  softmax_kernel<<<dim3(B_), 1024, 0, stream>>>(scores, laf, weights);
  kv_kernel<<<dim3(NSEG, B_), 256, 0, stream>>>(x, weights, kvPart, sPart);
  const size_t nred = (size_t)B_ * D_ * D_ + (size_t)B_ * D_;
  reduce_kernel<<<dim3((unsigned)((nred + 255) / 256)), 256, 0, stream>>>(kvPart, sPart, KVp, S);
  attend_kernel<<<dim3(B_ * (N_ / 64)), 128, 0, stream>>>(x, weights, KVp, S, attended);
}